// HybridNATModel_65481071399781
// MI455X (gfx1250) — compile-verified
//
#include <hip/hip_runtime.h>
#include <math.h>

typedef __attribute__((ext_vector_type(2))) float v2f;
typedef __attribute__((ext_vector_type(8))) float v8f;

#define BATCH 8192
#define EPS_BN 1e-5f

// ---------------------------------------------------------------------------
// Kernel 1: per-image fused  conv1+relu+pool (VALU)  ->  conv2 (WMMA f32
// 16x16x4 implicit GEMM) +relu+pool -> flat row, plus the three small heads.
// One workgroup (256 threads = 8 waves) per image.
// ---------------------------------------------------------------------------
__global__ __launch_bounds__(256) void k_fused_conv(
    const float* __restrict__ x,
    const float* __restrict__ c1w, const float* __restrict__ c1b,
    const float* __restrict__ c2w, const float* __restrict__ c2b,
    const float* __restrict__ samp_w1, const float* __restrict__ samp_b1,
    const float* __restrict__ samp_w2, const float* __restrict__ samp_b2,
    const float* __restrict__ est_w1, const float* __restrict__ est_b1,
    const float* __restrict__ est_w2, const float* __restrict__ est_b2,
    const float* __restrict__ est_w3, const float* __restrict__ est_b3,
    const float* __restrict__ memory,
    float* __restrict__ flat_g,
    float* __restrict__ out_probs, float* __restrict__ out_reg,
    float* __restrict__ out_kern)
{
    __shared__ float sx[900];     // padded 30x30 input
    __shared__ float sw1[72];     // conv1 weights (8 x 9)
    __shared__ float sb1[8];
    __shared__ float sw2[1152];   // conv2 weights transposed: [k=72][n=16]
    __shared__ int   offs[72];    // k -> LDS offset inside padded h1
    __shared__ float h1[2048];    // padded 8 x 16 x 16 (1-px zero border)
    __shared__ float h2[3136];    // conv2 out, layout [p=196][n=16]
    __shared__ float fl[784];     // pooled flat row

    const int img = blockIdx.x;
    const int tid = threadIdx.x;

    // ---- phase 1: stage everything in LDS ----
    for (int i = tid; i < 900; i += 256) {
        int yy = i / 30, xx = i % 30;
        float v = 0.f;
        if (yy >= 1 && yy <= 28 && xx >= 1 && xx <= 28)
            v = x[img * 784 + (yy - 1) * 28 + (xx - 1)];
        sx[i] = v;
    }
    for (int i = tid; i < 1152; i += 256) {      // (16,8,3,3) -> [k][n]
        int n = i / 72, k = i % 72;
        sw2[k * 16 + n] = c2w[i];
    }
    for (int i = tid; i < 2048; i += 256) h1[i] = 0.f;
    if (tid < 72) {
        sw1[tid]  = c1w[tid];
        offs[tid] = (tid / 9) * 256 + ((tid % 9) / 3) * 16 + (tid % 3);
    }
    if (tid < 8) sb1[tid] = c1b[tid];
    __syncthreads();

    // ---- conv1 + relu + maxpool2 -> h1 (padded) ----
    for (int i = tid; i < 1568; i += 256) {       // 8 ch * 14 * 14
        int ci = i / 196, p = i % 196, py = p / 14, px = p % 14;
        const float  bias = sb1[ci];
        const float* w = &sw1[ci * 9];
        float m = 0.f;                            // relu => max with 0 is free
        for (int sy = 0; sy < 2; ++sy)
        for (int sp = 0; sp < 2; ++sp) {
            int oy = 2 * py + sy, ox = 2 * px + sp;
            float acc = bias;
            #pragma unroll
            for (int dy = 0; dy < 3; ++dy)
                #pragma unroll
                for (int dx = 0; dx < 3; ++dx)
                    acc += sx[(oy + dy) * 30 + ox + dx] * w[dy * 3 + dx];
            m = fmaxf(m, acc);
        }
        h1[ci * 256 + (py + 1) * 16 + (px + 1)] = m;
    }
    __syncthreads();

    // ---- conv2 as implicit GEMM:  D(196x16) = A(196x72) * B(72x16) ----
    // WMMA f32 16x16x4, 13 M-tiles x 18 K-steps, wave-uniform control flow.
    const int wave = tid >> 5, lane = tid & 31;
    const int lm = lane & 15, hi = lane >> 4;
    for (int mt = wave; mt < 13; mt += 8) {
        int p = mt * 16 + lm;
        if (p > 195) p = 195;                     // clamp tail rows (discarded)
        const int base = (p / 14) * 16 + (p % 14); // padded h1 (oy+dy)(ox+dx)
        v8f acc = {};
        #pragma unroll
        for (int ks = 0; ks < 18; ++ks) {
            const int k0 = ks * 4 + hi * 2;
            v2f a, b;
            a.x = h1[offs[k0]     + base];
            a.y = h1[offs[k0 + 1] + base];
            b.x = sw2[k0 * 16 + lm];
            b.y = sw2[(k0 + 1) * 16 + lm];
            acc = __builtin_amdgcn_wmma_f32_16x16x4_f32(
                false, a, false, b, (short)0, acc, false, false);
        }
        const float bias = c2b[lm];
        #pragma unroll
        for (int v = 0; v < 8; ++v) {
            int pv = mt * 16 + v + hi * 8;        // C layout: M = v + 8*hi
            if (pv < 196) h2[pv * 16 + lm] = fmaxf(acc[v] + bias, 0.f);
        }
    }
    __syncthreads();

    // ---- maxpool2 -> flat (LDS + global) ----
    for (int i = tid; i < 784; i += 256) {        // index = n*49 + py*7 + px
        int n = i / 49, r = i % 49, py = r / 7, px = r % 7;
        int p00 = (2 * py) * 14 + 2 * px;
        float v = fmaxf(fmaxf(h2[p00 * 16 + n],        h2[(p00 + 1) * 16 + n]),
                        fmaxf(h2[(p00 + 14) * 16 + n], h2[(p00 + 15) * 16 + n]));
        fl[i] = v;
        flat_g[img * 784 + i] = v;
    }
    __syncthreads();

    // ---- heads (tiny, a few lanes) ----
    if (tid < 10) {                               // RBF kernel row, GAMMA=1
        float sq = 0.f;
        #pragma unroll
        for (int d = 0; d < 4; ++d) {
            float df = fl[d] - memory[tid * 4 + d];
            sq += df * df;
        }
        out_kern[img * 10 + tid] = expf(-sq);
    } else if (tid == 10) {                       // sampler + softmax(2)
        float f0 = fl[0], f1 = fl[1];
        float s0 = samp_b2[0], s1 = samp_b2[1];
        #pragma unroll
        for (int i = 0; i < 4; ++i) {
            float t = tanhf(f0 * samp_w1[i * 2] + f1 * samp_w1[i * 2 + 1] + samp_b1[i]);
            s0 += t * samp_w2[i];
            s1 += t * samp_w2[4 + i];
        }
        float mx = fmaxf(s0, s1);
        float e0 = expf(s0 - mx), e1 = expf(s1 - mx);
        float inv = 1.f / (e0 + e1);
        out_probs[img * 2 + 0] = e0 * inv;
        out_probs[img * 2 + 1] = e1 * inv;
    } else if (tid == 11) {                       // estimator MLP 2->8->4->1
        float f0 = fl[0], f1 = fl[1];
        float t1[8];
        #pragma unroll
        for (int i = 0; i < 8; ++i)
            t1[i] = tanhf(f0 * est_w1[i * 2] + f1 * est_w1[i * 2 + 1] + est_b1[i]);
        float t2[4];
        #pragma unroll
        for (int i = 0; i < 4; ++i) {
            float a = est_b2[i];
            #pragma unroll
            for (int j = 0; j < 8; ++j) a += t1[j] * est_w2[i * 8 + j];
            t2[i] = tanhf(a);
        }
        float r = est_b3[0];
        #pragma unroll
        for (int i = 0; i < 4; ++i) r += t2[i] * est_w3[i];
        out_reg[img] = r;
    }
}

// ---------------------------------------------------------------------------
// Kernel 2: fc1 GEMM via WMMA f32 16x16x4.  C(8192x64)=flat(8192x784)@W^T,
// +bias, relu.  One 16x16 C tile per wave, 196 chained WMMAs.
// ---------------------------------------------------------------------------
__global__ __launch_bounds__(256) void k_fc1_wmma(
    const float* __restrict__ flat, const float* __restrict__ fc1_w,
    const float* __restrict__ fc1_b, float* __restrict__ fc1o)
{
    const int wave = threadIdx.x >> 5, lane = threadIdx.x & 31;
    const int lm = lane & 15, hi = lane >> 4;
    const int gw = blockIdx.x * 8 + wave;         // 0..2047
    const int mt = gw >> 2, nt = gw & 3;          // 512 row tiles x 4 col tiles

    const float* Ap = flat  + (size_t)(mt * 16 + lm) * 784 + hi * 2;
    const float* Bp = fc1_w + (size_t)(nt * 16 + lm) * 784 + hi * 2; // B[k][n]=W[n][k]
    v8f acc = {};
    #pragma unroll 4
    for (int k = 0; k < 784; k += 4) {
        v2f a = *(const v2f*)(Ap + k);            // 8B aligned
        v2f b = *(const v2f*)(Bp + k);
        acc = __builtin_amdgcn_wmma_f32_16x16x4_f32(
            false, a, false, b, (short)0, acc, false, false);
    }
    const int col = nt * 16 + lm;
    const float bias = fc1_b[col];
    #pragma unroll
    for (int v = 0; v < 8; ++v) {
        int row = mt * 16 + v + hi * 8;
        fc1o[(size_t)row * 64 + col] = fmaxf(acc[v] + bias, 0.f);
    }
}

// ---------------------------------------------------------------------------
// Kernel 3: fc2 (64 -> 4), one thread per output element.
// ---------------------------------------------------------------------------
__global__ __launch_bounds__(256) void k_fc2(
    const float* __restrict__ fc1o, const float* __restrict__ w,
    const float* __restrict__ b, float* __restrict__ fc)
{
    int i = blockIdx.x * blockDim.x + threadIdx.x;
    if (i >= BATCH * 4) return;
    int row = i >> 2, o = i & 3;
    const float* xr = fc1o + (size_t)row * 64;
    const float* wr = w + o * 64;
    float a = b[o];
    #pragma unroll 8
    for (int j = 0; j < 64; ++j) a += xr[j] * wr[j];
    fc[i] = a;
}

// ---------------------------------------------------------------------------
// Kernel 4: deterministic batch mean/var (ddof=0) + normalize -> logits.
// Single workgroup, tree reduction (no float atomics => replay-stable).
// ---------------------------------------------------------------------------
__global__ __launch_bounds__(256) void k_bn(
    const float* __restrict__ fc, const float* __restrict__ g,
    const float* __restrict__ bb, float* __restrict__ logits)
{
    __shared__ float s_sum[4 * 256];
    __shared__ float s_sq[4 * 256];
    __shared__ float s_mu[4], s_inv[4], s_beta[4];
    const int tid = threadIdx.x;

    float sum[4] = {0.f, 0.f, 0.f, 0.f}, sq[4] = {0.f, 0.f, 0.f, 0.f};
    for (int r = tid; r < BATCH; r += 256) {
        #pragma unroll
        for (int d = 0; d < 4; ++d) {
            float v = fc[r * 4 + d];
            sum[d] += v; sq[d] += v * v;
        }
    }
    #pragma unroll
    for (int d = 0; d < 4; ++d) { s_sum[d * 256 + tid] = sum[d]; s_sq[d * 256 + tid] = sq[d]; }
    __syncthreads();
    for (int off = 128; off > 0; off >>= 1) {
        if (tid < off) {
            #pragma unroll
            for (int d = 0; d < 4; ++d) {
                s_sum[d * 256 + tid] += s_sum[d * 256 + tid + off];
                s_sq[d * 256 + tid]  += s_sq[d * 256 + tid + off];
            }
        }
        __syncthreads();
    }
    if (tid < 4) {
        float mu  = s_sum[tid * 256] * (1.f / BATCH);
        float var = s_sq[tid * 256] * (1.f / BATCH) - mu * mu;
        s_mu[tid]   = mu;
        s_inv[tid]  = g[tid] * rsqrtf(var + EPS_BN);
        s_beta[tid] = bb[tid];
    }
    __syncthreads();
    for (int i = tid; i < BATCH * 4; i += 256) {
        int d = i & 3;
        logits[i] = (fc[i] - s_mu[d]) * s_inv[d] + s_beta[d];
    }
}

// ---------------------------------------------------------------------------
extern "C" void kernel_launch(void* const* d_in, const int* in_sizes, int n_in,
                              void* d_out, int out_size, void* d_ws, size_t ws_size,
                              hipStream_t stream)
{
    (void)in_sizes; (void)n_in; (void)out_size; (void)ws_size;
    const float* x       = (const float*)d_in[0];
    const float* conv1_w = (const float*)d_in[1];
    const float* conv1_b = (const float*)d_in[2];
    const float* conv2_w = (const float*)d_in[3];
    const float* conv2_b = (const float*)d_in[4];
    const float* fc1_w   = (const float*)d_in[5];
    const float* fc1_b   = (const float*)d_in[6];
    const float* fc2_w   = (const float*)d_in[7];
    const float* fc2_b   = (const float*)d_in[8];
    const float* bn_g    = (const float*)d_in[9];
    const float* bn_b    = (const float*)d_in[10];
    const float* samp_w1 = (const float*)d_in[11];
    const float* samp_b1 = (const float*)d_in[12];
    const float* samp_w2 = (const float*)d_in[13];
    const float* samp_b2 = (const float*)d_in[14];
    const float* est_w1  = (const float*)d_in[15];
    const float* est_b1  = (const float*)d_in[16];
    const float* est_w2  = (const float*)d_in[17];
    const float* est_b2  = (const float*)d_in[18];
    const float* est_w3  = (const float*)d_in[19];
    const float* est_b3  = (const float*)d_in[20];
    const float* memory  = (const float*)d_in[21];

    // workspace layout (floats): flat[8192*784] | fc1out[8192*64] | fc[8192*4]
    float* ws    = (float*)d_ws;
    float* flat  = ws;                             // 6,422,528 f
    float* fc1o  = ws + 6422528;                   //   524,288 f
    float* fc    = ws + 6946816;                   //    32,768 f  (~27.9 MB total)

    // output layout: logits(8192x4) | probs(8192x2) | reg(8192x1) | kern(8192x10)
    float* logits = (float*)d_out;
    float* probs  = logits + BATCH * 4;
    float* reg    = logits + BATCH * 6;
    float* kern   = logits + BATCH * 7;

    k_fused_conv<<<BATCH, 256, 0, stream>>>(
        x, conv1_w, conv1_b, conv2_w, conv2_b,
        samp_w1, samp_b1, samp_w2, samp_b2,
        est_w1, est_b1, est_w2, est_b2, est_w3, est_b3,
        memory, flat, probs, reg, kern);

    k_fc1_wmma<<<256, 256, 0, stream>>>(flat, fc1_w, fc1_b, fc1o);
    k_fc2<<<(BATCH * 4 + 255) / 256, 256, 0, stream>>>(fc1o, fc2_w, fc2_b, fc);
    k_bn<<<1, 256, 0, stream>>>(fc, bn_g, bn_b, logits);
}